// DRL_22162031247575
// MI455X (gfx1250) — compile-verified
//
#include <hip/hip_runtime.h>
#include <stdint.h>

// ---------------------------------------------------------------------------
// DRL state-update: pure data movement (~230 MB traffic, ~0 FLOPs).
// Roofline @ 23.3 TB/s HBM ~= 10 us. No matmul -> no WMMA; the gfx1250 paths
// that matter are async LDS DMA (ASYNCcnt) and NT cache hints (th:*_NT).
// ---------------------------------------------------------------------------

typedef float v4f __attribute__((ext_vector_type(4)));
typedef int   v4i __attribute__((ext_vector_type(4)));

#define AS1 __attribute__((address_space(1)))
#define AS3 __attribute__((address_space(3)))

// Probe for gfx1250 async global<->LDS builtins (ISA: GLOBAL_LOAD_ASYNC_TO_LDS_B128).
#if defined(__has_builtin)
#if __has_builtin(__builtin_amdgcn_global_load_async_to_lds_b128) && \
    __has_builtin(__builtin_amdgcn_global_store_async_from_lds_b128) && \
    __has_builtin(__builtin_amdgcn_s_wait_asynccnt)
#define USE_ASYNC_LDS 1
#endif
#endif

// Problem constants (BS=512, MC=128, K=200, E=128, T=64)
static constexpr int N_MASKFS_V4 = 512 * 128 * 50;   // 3,276,800 float4 groups
static constexpr int BLOCKS_A    = N_MASKFS_V4 / 256; // 12800
static constexpr int COPY_UNITS  = 512 * 4160;        // 2,129,920 float4 units
static constexpr int BLOCKS_B    = COPY_UNITS / 256;  // 8320
static constexpr int BLOCKS_C    = 512;               // one per batch
static constexpr int GRID        = BLOCKS_A + BLOCKS_B + BLOCKS_C; // 21632

// Flat float offsets of each output in d_out (reference return order)
static constexpr int OFF_MASKFS = 0;          // (512,128,200)
static constexpr int OFF_S      = 13107200;   // (512,1,400)
static constexpr int OFF_SH     = 13312000;   // (512,1,128)
static constexpr int OFF_D      = 13377536;   // (512,64,1)
static constexpr int OFF_MF     = 13410304;   // (512,64,1)
static constexpr int OFF_MFS    = 13443072;   // (512,1,200)
static constexpr int OFF_DYN    = 13545472;   // (512,65,128)
static constexpr int OFF_MFN    = 17805312;   // (512,65,128)

__global__ __launch_bounds__(256) void drl_fused_kernel(
    const float* __restrict__ st,   // static   (512,128,400)
    const float* __restrict__ sth,  // static_h (512,128,128)
    const float* __restrict__ dyn,  // dynamic  (512,64,128)
    const float* __restrict__ mf,   // mask_f   (512,64,128)
    const int* __restrict__ cs,     // couriers_selected (512,1)
    const int* __restrict__ ts,     // sensingtask_selected (512,1)
    float* __restrict__ out)
{
    const float NEG_INF = -__builtin_inff();
    const int tid = threadIdx.x;
    const int blk = blockIdx.x;

    __shared__ v4f stage[256]; // 4KB staging slot per block for async LDS DMA

    if (blk < BLOCKS_A) {
        // ---- Region A: mask_fs = where(pair==0, -inf, 0), col k=0 -> -inf ----
        // One lane: 4 pairs (8 floats = 2 x b128 NT loads) -> 1 x b128 NT store.
        int gid = blk * 256 + tid;          // float4-group index in mask_fs
        int row = gid / 50;                 // b*128 + m
        int g   = gid - row * 50;           // group within row (0..49)
        const v4f* src = (const v4f*)(st + row * 400 + g * 8);
        v4f a = __builtin_nontemporal_load(src);
        v4f b = __builtin_nontemporal_load(src + 1);
        v4f o;
        o.x = (a.x == 0.0f && a.y == 0.0f) ? NEG_INF : 0.0f;
        o.y = (a.z == 0.0f && a.w == 0.0f) ? NEG_INF : 0.0f;
        o.z = (b.x == 0.0f && b.y == 0.0f) ? NEG_INF : 0.0f;
        o.w = (b.z == 0.0f && b.w == 0.0f) ? NEG_INF : 0.0f;
        if (g == 0) o.x = NEG_INF;          // mask_fs[:,:,0] = -inf
        __builtin_nontemporal_store(o, (v4f*)(out + OFF_MASKFS + row * 200 + g * 4));
    } else if (blk < BLOCKS_A + BLOCKS_B) {
        // ---- Region B: dynamic_new / mask_f_new (copy rows 0..63 + build row 64) ----
        int cid   = (blk - BLOCKS_A) * 256 + tid;   // [0, 512*4160)
        int batch = cid / 4160;
        int r     = cid - batch * 4160;             // 0..4159
        int c     = cs[batch];
        bool isDyn = (r < 2080);
        int rr     = isDyn ? r : (r - 2080);        // 0..2079 within tensor
        const float* srcBase = isDyn ? dyn : mf;
        float* dstBase = out + (isDyn ? OFF_DYN : OFF_MFN) + batch * 8320;
        if (rr < 2048) {
            // Bulk copy: 16B per lane. Prefer async LDS DMA (global->LDS->global,
            // never touches VGPRs, tracked by ASYNCcnt).
            const v4f* s4 = (const v4f*)(srcBase + batch * 8192 + rr * 4);
            v4f* d4 = (v4f*)(dstBase + rr * 4);
#if defined(USE_ASYNC_LDS)
            AS3 v4i* lp = (AS3 v4i*)(uint32_t)(uintptr_t)(&stage[tid]);
            __builtin_amdgcn_global_load_async_to_lds_b128(
                (AS1 v4i*)(uintptr_t)s4, lp, 0, 0);
            __builtin_amdgcn_s_wait_asynccnt(0);
            __builtin_amdgcn_global_store_async_from_lds_b128(
                (AS1 v4i*)(uintptr_t)d4, lp, 0, 0);
            // outstanding ASYNCcnt drained by implicit wait-idle at s_endpgm
#else
            v4f t = __builtin_nontemporal_load(s4);
            __builtin_nontemporal_store(t, d4);
#endif
        } else {
            // Row 64: add_state / add_mask_f synthesized in registers.
            int j   = rr - 2048;   // 0..31 (float4 groups across MC=128)
            int col = j * 4;
            v4f o;
            if (isDyn) {
                float task = (float)ts[batch];
                o.x = (col + 0 == c) ? task : 0.0f;
                o.y = (col + 1 == c) ? task : 0.0f;
                o.z = (col + 2 == c) ? task : 0.0f;
                o.w = (col + 3 == c) ? task : 0.0f;
            } else {
                o.x = (col + 0 == c) ? 0.0f : NEG_INF;
                o.y = (col + 1 == c) ? 0.0f : NEG_INF;
                o.z = (col + 2 == c) ? 0.0f : NEG_INF;
                o.w = (col + 3 == c) ? 0.0f : NEG_INF;
            }
            __builtin_nontemporal_store(o, (v4f*)(dstBase + 8192 + j * 4));
        }
    } else {
        // ---- Region C: per-batch gathers (s, sh, d, mf, mfs) ----
        int b = blk - (BLOCKS_A + BLOCKS_B);
        int c = cs[b];
        const float* srow  = st  + (b * 128 + c) * 400;
        const float* shrow = sth + (b * 128 + c) * 128;
        for (int i = tid; i < 400; i += 256)
            out[OFF_S + b * 400 + i] = srow[i];
        for (int i = tid; i < 128; i += 256)
            out[OFF_SH + b * 128 + i] = shrow[i];
        for (int i = tid; i < 64; i += 256)
            out[OFF_D + b * 64 + i] = dyn[(b * 64 + i) * 128 + c];
        for (int i = tid; i < 64; i += 256)
            out[OFF_MF + b * 64 + i] = mf[(b * 64 + i) * 128 + c];
        for (int i = tid; i < 200; i += 256) {
            float a0 = srow[2 * i], a1 = srow[2 * i + 1];
            float v = (a0 == 0.0f && a1 == 0.0f) ? NEG_INF : 0.0f;
            if (i == 0) v = NEG_INF;        // mask_fs[:,:,0]
            out[OFF_MFS + b * 200 + i] = v;
        }
    }
}

extern "C" void kernel_launch(void* const* d_in, const int* in_sizes, int n_in,
                              void* d_out, int out_size, void* d_ws, size_t ws_size,
                              hipStream_t stream) {
    (void)in_sizes; (void)n_in; (void)out_size; (void)d_ws; (void)ws_size;
    const float* st  = (const float*)d_in[0];  // static
    const float* sth = (const float*)d_in[1];  // static_h
    const float* dyn = (const float*)d_in[2];  // dynamic
    const float* mfp = (const float*)d_in[3];  // mask_f
    const int*   cs  = (const int*)d_in[4];    // couriers_selected
    const int*   ts  = (const int*)d_in[5];    // sensingtask_selected
    float* out = (float*)d_out;
    hipLaunchKernelGGL(drl_fused_kernel, dim3(GRID), dim3(256), 0, stream,
                       st, sth, dyn, mfp, cs, ts, out);
}